// RegionSelector_72533407695352
// MI455X (gfx1250) — compile-verified
//
#include <hip/hip_runtime.h>
#include <math.h>

// MI455X / gfx1250 (CDNA5, wave32). Bandwidth-bound streaming reduction:
//   201 MB read / 23.3 TB/s  =>  ~8.6 us floor. Everything else is noise.
// Kernel 1: one block per (batch, 4x4 grid cell); 256 threads stream the
//           3x128x128 block with coalesced NT b128 loads, then reduce the
//           wave via two chained V_WMMA_F32_16X16X4_F32 (row-sum vs ones).
// Kernel 2: per-batch 3x3 window sums over the 16 cell sums + argmax.

typedef __attribute__((ext_vector_type(2))) float v2f;
typedef __attribute__((ext_vector_type(4))) float v4f;
typedef __attribute__((ext_vector_type(8))) float v8f;

#define NBATCH 64
#define NCH    3
#define HH     512
#define WW     512
#define CELL   128           // 512 / GRID_SIZE(4)
#define CHUNKS (NCH * CELL * (CELL / 4))   // 12288 b128 loads per block

__device__ __forceinline__ float wave32_sum_wmma(float acc) {
    // WMMA #1: A[16x4] = lanes' partials (a0 = acc, a1 = 0), B = ones(4x16).
    // D[m][*] = acc[m] + acc[m+16]  (rows 0-7 -> d0..d7 of lanes 0-15,
    //                                rows 8-15 -> d0..d7 of lanes 16-31).
    v2f a;   a[0] = acc;  a[1] = 0.0f;
    v2f one; one[0] = 1.0f; one[1] = 1.0f;
    v8f cz = {0.f, 0.f, 0.f, 0.f, 0.f, 0.f, 0.f, 0.f};
    v8f d1 = __builtin_amdgcn_wmma_f32_16x16x4_f32(
        false, a, false, one, (short)0, cz, false, false);

    // Fold the 8 row-sums into a new A and row-sum again: every lane now
    // holds the full 32-lane sum in every D register.
    v2f a2;
    a2[0] = (d1[0] + d1[1]) + (d1[2] + d1[3]);
    a2[1] = (d1[4] + d1[5]) + (d1[6] + d1[7]);
    v8f d2 = __builtin_amdgcn_wmma_f32_16x16x4_f32(
        false, a2, false, one, (short)0, cz, false, false);
    return d2[0];
}

__global__ __launch_bounds__(256) void cell_sums_kernel(
        const float* __restrict__ x, float* __restrict__ cell_sums) {
    const int blk = blockIdx.x;        // b*16 + gr*4 + gc
    const int b   = blk >> 4;
    const int gr  = (blk >> 2) & 3;
    const int gc  = blk & 3;

    const float* base = x + (size_t)b * NCH * HH * WW
                          + (size_t)gr * CELL * WW
                          + (size_t)gc * CELL;

    // 48 fully-coalesced b128 NT loads per thread; uniform trip count so
    // EXEC stays all-1s for the WMMAs below.
    float acc = 0.0f;
    #pragma unroll 4
    for (int i = threadIdx.x; i < CHUNKS; i += 256) {
        const int ch   = i >> 12;      // / 4096
        const int rem  = i & 4095;
        const int row  = rem >> 5;     // / 32
        const int col4 = rem & 31;
        const v4f* p = (const v4f*)(base + (size_t)ch * HH * WW
                                         + (size_t)row * WW + col4 * 4);
        const v4f v = __builtin_nontemporal_load(p);
        acc += (v[0] + v[1]) + (v[2] + v[3]);
    }

    const float wsum = wave32_sum_wmma(acc);

    __shared__ float lds[8];
    const int lane = threadIdx.x & 31;
    const int wid  = threadIdx.x >> 5;
    if (lane == 0) lds[wid] = wsum;
    __syncthreads();
    if (threadIdx.x == 0) {
        float s = 0.0f;
        #pragma unroll
        for (int i = 0; i < 8; ++i) s += lds[i];
        cell_sums[blk] = s;
    }
}

__global__ __launch_bounds__(64) void select_kernel(
        const float* __restrict__ cell_sums, float* __restrict__ out) {
    const int b = blockIdx.x * blockDim.x + threadIdx.x;
    if (b >= NBATCH) return;

    float c[16];
    #pragma unroll
    for (int i = 0; i < 16; ++i) c[i] = cell_sums[b * 16 + i];

    // 3x3 valid windows over the 4x4 cell grid -> 4 candidates.
    float best = -INFINITY;
    int bestIdx = 0;
    #pragma unroll
    for (int r = 0; r < 2; ++r) {
        #pragma unroll
        for (int col = 0; col < 2; ++col) {
            float s = 0.0f;
            #pragma unroll
            for (int i = 0; i < 3; ++i)
                #pragma unroll
                for (int j = 0; j < 3; ++j)
                    s += c[(r + i) * 4 + (col + j)];
            const int idx = r * 2 + col;
            if (s > best) { best = s; bestIdx = idx; }   // strict '>' keeps
        }                                                // lowest index on tie
    }
    out[b * 2 + 0] = (float)(bestIdx >> 1);  // row
    out[b * 2 + 1] = (float)(bestIdx & 1);   // col
}

extern "C" void kernel_launch(void* const* d_in, const int* in_sizes, int n_in,
                              void* d_out, int out_size, void* d_ws, size_t ws_size,
                              hipStream_t stream) {
    (void)in_sizes; (void)n_in; (void)out_size; (void)ws_size;
    const float* x = (const float*)d_in[0];
    float* cell_sums = (float*)d_ws;      // 64 * 16 floats = 4 KB scratch
    float* out = (float*)d_out;           // 64 * 1 * 2 coords

    cell_sums_kernel<<<NBATCH * 16, 256, 0, stream>>>(x, cell_sums);
    select_kernel<<<1, 64, 0, stream>>>(cell_sums, out);
}